// Extractor_multimodal_32272384262540
// MI455X (gfx1250) — compile-verified
//
#include <hip/hip_runtime.h>
#include <math.h>

// ---------------------------------------------------------------------------
// Problem constants (from reference): B=4, H=W=64, N=4096, C=768,
// HEADS=6, POINTS=4, HID=192, HD=128.  M = B*N = 16384 rows everywhere.
// ---------------------------------------------------------------------------
#define BB      4
#define HH      64
#define WW      64
#define NN      4096            // H*W
#define CC      768
#define HEADS   6
#define POINTS  4
#define HID     192
#define HD      128
#define MROWS   (BB * NN)       // 16384

#define NT      3               // 16-wide N subtiles per block (48 columns)
#define KC      192             // K staging chunk (divides 768 and 192)

typedef __attribute__((ext_vector_type(2))) float v2f;
typedef __attribute__((ext_vector_type(8))) float v8f;

// ---------------------------------------------------------------------------
// LayerNorm over C=768.  One 256-thread block per row; 3 elements per thread.
// ---------------------------------------------------------------------------
__global__ __launch_bounds__(256)
void ln768_kernel(const float* __restrict__ x, const float* __restrict__ scale,
                  const float* __restrict__ bias, float* __restrict__ y)
{
    const int row  = blockIdx.x;
    const int tid  = threadIdx.x;
    const int lane = tid & 31;
    const int wid  = tid >> 5;

    const float* xr = x + (size_t)row * CC;
    float v0 = xr[tid], v1 = xr[tid + 256], v2 = xr[tid + 512];
    float s = v0 + v1 + v2;
    float q = v0 * v0 + v1 * v1 + v2 * v2;
    #pragma unroll
    for (int o = 16; o > 0; o >>= 1) {
        s += __shfl_down(s, o, 32);
        q += __shfl_down(q, o, 32);
    }
    __shared__ float rs[8], rq[8], stats[2];
    if (lane == 0) { rs[wid] = s; rq[wid] = q; }
    __syncthreads();
    if (tid == 0) {
        float ts = 0.f, tq = 0.f;
        #pragma unroll
        for (int i = 0; i < 8; ++i) { ts += rs[i]; tq += rq[i]; }
        float mu  = ts * (1.0f / CC);
        float var = tq * (1.0f / CC) - mu * mu;
        stats[0] = mu;
        stats[1] = rsqrtf(var + 1e-6f);
    }
    __syncthreads();
    const float mu = stats[0], rstd = stats[1];
    float* yr = y + (size_t)row * CC;
    yr[tid]       = (v0 - mu) * rstd * scale[tid]       + bias[tid];
    yr[tid + 256] = (v1 - mu) * rstd * scale[tid + 256] + bias[tid + 256];
    yr[tid + 512] = (v2 - mu) * rstd * scale[tid + 512] + bias[tid + 512];
}

// ---------------------------------------------------------------------------
// GEMM: out[M,N] = alpha * (A[M,K] @ W[K,N] + bias[N]) + res[M,N]
// fp32 WMMA 16x16x4.  One wave computes a 16x48 strip (3 accumulators) so the
// A fragment loaded from global is reused for 3 WMMAs (3x less L2 traffic).
// 8 waves/block cover 128 rows; the block stages NT K-chunk weight tiles
// (3 x 192 x 16 floats = 36 KB) in LDS, chunked over K with prefetch of the
// next A chunk overlapping the staging barrier.
//
// fp32 A 16x4 frag: lane<16 -> {A[m=l][k], A[m=l][k+1]},
//                   lane>=16 -> {A[m=l-16][k+2], A[m=l-16][k+3]}
// fp32 B 4x16 frag: lane<16 -> {B[k][n=l], B[k+1][n=l]},
//                   lane>=16 -> {B[k+2][n=l-16], B[k+3][n=l-16]}
// C/D: VGPR v, lane l:  D[m = v + 8*(l>=16)][n = l&15]
// ---------------------------------------------------------------------------
__global__ __launch_bounds__(256)
void wmma_gemm_kernel(const float* __restrict__ A, const float* __restrict__ W,
                      const float* __restrict__ bias, const float* __restrict__ res,
                      float* __restrict__ out, int M, int N, int K, float alpha)
{
    __shared__ float wlds[NT * KC * 16];     // 36 KB
    const int tid   = threadIdx.x;
    const int lane  = tid & 31;
    const int wid   = tid >> 5;
    const int nBase = blockIdx.x * (16 * NT);
    const int mBase = (blockIdx.y * 8 + wid) * 16;

    const int half = lane >> 4;              // 0: K=0,1   1: K=2,3
    const int l16  = lane & 15;
    const float* arow = A + (size_t)(mBase + l16) * K + 2 * half;
    __builtin_prefetch(arow, 0, 3);          // global_prefetch: warm first A chunk

    const float* w0 = wlds + l16 + 32 * half;              // subtile 0
    const float* w1 = w0 + KC * 16;                        // subtile 1
    const float* w2 = w1 + KC * 16;                        // subtile 2

    v8f c0 = {}, c1 = {}, c2 = {};

    for (int k0 = 0; k0 < K; k0 += KC) {
        // Cooperative stage of NT weight chunk-tiles (zero-pad cols >= N).
        for (int e = tid; e < NT * KC * 16; e += 256) {
            const int t   = e / (KC * 16);
            const int r   = e - t * (KC * 16);
            const int k   = r >> 4, c16 = r & 15;
            const int col = nBase + t * 16 + c16;
            wlds[e] = (col < N) ? W[(size_t)(k0 + k) * N + col] : 0.f;
        }
        if (k0 + KC < K) __builtin_prefetch(arow + k0 + KC, 0, 3);
        __syncthreads();

        const float* ak = arow + k0;
        #pragma unroll 4
        for (int k = 0; k < KC; k += 4) {
            v2f a;
            const float2 av = *(const float2*)(ak + k);    // A[m][k+2h .. k+2h+1]
            a.x = av.x; a.y = av.y;
            v2f b0, b1, b2;
            b0.x = w0[k * 16]; b0.y = w0[k * 16 + 16];
            b1.x = w1[k * 16]; b1.y = w1[k * 16 + 16];
            b2.x = w2[k * 16]; b2.y = w2[k * 16 + 16];
            c0 = __builtin_amdgcn_wmma_f32_16x16x4_f32(false, a, false, b0,
                                                       (short)0, c0, false, false);
            c1 = __builtin_amdgcn_wmma_f32_16x16x4_f32(false, a, false, b1,
                                                       (short)0, c1, false, false);
            c2 = __builtin_amdgcn_wmma_f32_16x16x4_f32(false, a, false, b2,
                                                       (short)0, c2, false, false);
        }
        __syncthreads();
    }

    // Epilogue: bias, alpha-scale, optional residual, guarded stores.
    #pragma unroll
    for (int t = 0; t < NT; ++t) {
        const int col = nBase + t * 16 + l16;
        if (col >= N) continue;
        const v8f cc = (t == 0) ? c0 : (t == 1) ? c1 : c2;
        const float bcol = bias ? bias[col] : 0.f;
        #pragma unroll
        for (int v = 0; v < 8; ++v) {
            const int row = mBase + half * 8 + v;
            float val = alpha * (cc[v] + bcol);
            if (res) val += res[(size_t)row * N + col];
            out[(size_t)row * N + col] = val;
        }
    }
}

// ---------------------------------------------------------------------------
// Deformable-attention sampling.  One wave per (b, n, head); 4 channels/lane.
// out[b,n, h*HD + 4*lane .. +3] = sum_p softmax(aw)_p * bilinear(v, loc_p)
// Corner validity & indices are wave-uniform -> no divergence.
// ---------------------------------------------------------------------------
__device__ __forceinline__
void sample_corner(float4& acc, const float* __restrict__ vb, int xi, int yi, float w)
{
    if (xi >= 0 && xi < WW && yi >= 0 && yi < HH) {
        const float4 t = *(const float4*)(vb + (size_t)(yi * WW + xi) * CC);
        acc.x += w * t.x; acc.y += w * t.y; acc.z += w * t.z; acc.w += w * t.w;
    }
}

__global__ __launch_bounds__(256)
void msda_sample_kernel(const float* __restrict__ v,   // [B, N, C] (values)
                        const float* __restrict__ off, // [B*N, 48]
                        const float* __restrict__ awl, // [B*N, 24] (logits)
                        const float* __restrict__ ref, // [B*N, 2]
                        float* __restrict__ out)       // [B, N, C]
{
    const int tid  = threadIdx.x;
    const int lane = tid & 31;
    const int wg   = blockIdx.x * 8 + (tid >> 5);   // global wave id
    const int h    = wg % HEADS;
    const int n    = (wg / HEADS) & (NN - 1);
    const int b    = wg / (HEADS * NN);
    const size_t bn = (size_t)b * NN + n;

    const float rx = ref[bn * 2 + 0];
    const float ry = ref[bn * 2 + 1];
    const float* offp = off + bn * (HEADS * POINTS * 2) + h * (POINTS * 2);
    const float* ap   = awl + bn * (HEADS * POINTS)     + h * POINTS;

    // softmax over the 4 points
    const float l0 = ap[0], l1 = ap[1], l2 = ap[2], l3 = ap[3];
    const float mx = fmaxf(fmaxf(l0, l1), fmaxf(l2, l3));
    const float e0 = expf(l0 - mx), e1 = expf(l1 - mx),
                e2 = expf(l2 - mx), e3 = expf(l3 - mx);
    const float inv = 1.f / (e0 + e1 + e2 + e3);
    const float wpt[POINTS] = { e0 * inv, e1 * inv, e2 * inv, e3 * inv };

    const int ch = h * HD + lane * 4;
    const float* vb = v + (size_t)b * NN * CC + ch;

    float4 acc = {0.f, 0.f, 0.f, 0.f};
    #pragma unroll
    for (int p = 0; p < POINTS; ++p) {
        // loc = ref + off/[W,H]; px = loc_x*W - 0.5  ==  rx*W + off_x - 0.5
        const float px = rx * (float)WW + offp[2 * p]     - 0.5f;
        const float py = ry * (float)HH + offp[2 * p + 1] - 0.5f;
        const float fx0 = floorf(px), fy0 = floorf(py);
        const float fx = px - fx0, fy = py - fy0;
        const int x0 = (int)fx0, y0 = (int)fy0;
        const float wp = wpt[p];
        sample_corner(acc, vb, x0,     y0,     (1.f - fx) * (1.f - fy) * wp);
        sample_corner(acc, vb, x0 + 1, y0,     fx         * (1.f - fy) * wp);
        sample_corner(acc, vb, x0,     y0 + 1, (1.f - fx) * fy         * wp);
        sample_corner(acc, vb, x0 + 1, y0 + 1, fx         * fy         * wp);
    }
    *(float4*)(out + bn * CC + ch) = acc;
}

// ---------------------------------------------------------------------------
// Depthwise 3x3 SAME conv (+bias) on [B,64,64,HID] followed by exact GELU.
// One thread per (b,y,x,c).
// ---------------------------------------------------------------------------
__global__ __launch_bounds__(256)
void dwconv_gelu_kernel(const float* __restrict__ x,   // [B, N, HID]
                        const float* __restrict__ wdw, // [3,3,1,HID]
                        const float* __restrict__ bdw, // [HID]
                        float* __restrict__ y)         // [B, N, HID]
{
    const int i = blockIdx.x * 256 + threadIdx.x;     // B*N*HID threads
    const int c  = i % HID;
    const int xx = (i / HID) & (WW - 1);
    const int yy = (i / (HID * WW)) & (HH - 1);
    const int b  = i / (HID * WW * HH);

    float acc = bdw[c];
    #pragma unroll
    for (int ky = -1; ky <= 1; ++ky) {
        const int sy = yy + ky;
        if (sy < 0 || sy >= HH) continue;
        #pragma unroll
        for (int kx = -1; kx <= 1; ++kx) {
            const int sx = xx + kx;
            if (sx < 0 || sx >= WW) continue;
            const float xv = x[((size_t)b * NN + sy * WW + sx) * HID + c];
            const float wv = wdw[((ky + 1) * 3 + (kx + 1)) * HID + c];
            acc = fmaf(xv, wv, acc);
        }
    }
    // exact GELU: 0.5 * x * (1 + erf(x / sqrt(2)))
    const float g = 0.5f * acc * (1.f + erff(acc * 0.70710678118654752f));
    y[i] = g;
}

// ---------------------------------------------------------------------------
// Host-side orchestration
// ---------------------------------------------------------------------------
static inline void launch_gemm(hipStream_t s, const float* A, const float* W,
                               const float* bias, const float* res, float* out,
                               int M, int N, int K, float alpha)
{
    dim3 grid((N + 16 * NT - 1) / (16 * NT), M / 128);
    wmma_gemm_kernel<<<grid, 256, 0, s>>>(A, W, bias, res, out, M, N, K, alpha);
}

extern "C" void kernel_launch(void* const* d_in, const int* in_sizes, int n_in,
                              void* d_out, int out_size, void* d_ws, size_t ws_size,
                              hipStream_t stream)
{
    (void)in_sizes; (void)n_in; (void)out_size; (void)ws_size;

    // Top-level inputs (setup_inputs() insertion order)
    const float* query = (const float*)d_in[0];   // [B,N,C]
    const float* qom   = (const float*)d_in[1];   // [3,B,N,C]
    const float* refp  = (const float*)d_in[2];   // [B,N,1,2]
    const float* feat  = (const float*)d_in[3];   // [B,N,C]
    // d_in[4] spatial_shapes, d_in[5] level_start_index, d_in[6] H, d_in[7] W
    // params pytree (sorted keys recursively) starts at index 8:
    //   depth(18), event(18), feat_norm{bias,scale}(2), lidar(18), rgb(18)
    // per-modality leaf order:
    //   +0 Waw +1 Woff +2 Wout +3 Wv +4 baw +5 boff +6 bout +7 bv
    //   +8 W1 +9 W2 +10 Wdw +11 b1 +12 b2 +13 bdw
    //   +14 ffn_norm.bias +15 ffn_norm.scale +16 qn.bias +17 qn.scale
    auto P = [&](int i) { return (const float*)d_in[i]; };
    const int BASE[4] = { 64, 8, 26, 46 };        // processing order: rgb, depth, event, lidar
    const float* fn_bias  = P(44);
    const float* fn_scale = P(45);

    // Workspace carve-up (floats)
    const size_t BNC = (size_t)MROWS * CC;        // 12.58M
    float* ws    = (float*)d_ws;
    float* featn = ws;                            // [B,N,C]  LN(feat)
    float* bufB  = ws + BNC;                      // qn, then sampled attention
    float* bufC  = ws + 2 * BNC;                  // values v, then LN for FFN
    float* offb  = ws + 3 * BNC;                  // [B*N,48]
    float* awb   = offb + (size_t)MROWS * (HEADS * POINTS * 2);  // [B*N,24]
    float* h1    = awb  + (size_t)MROWS * (HEADS * POINTS);      // [B*N,HID]
    float* hdw   = h1   + (size_t)MROWS * HID;                   // [B*N,HID]

    float* outq = (float*)d_out;                  // query' | qom'(x3), each [B,N,C]

    // Shared feature norm
    ln768_kernel<<<MROWS, 256, 0, stream>>>(feat, fn_scale, fn_bias, featn);

    for (int mi = 0; mi < 4; ++mi) {
        const int base = BASE[mi];
        const float* qsrc = (mi == 0) ? query : (qom + (size_t)(mi - 1) * BNC);
        float* qdst = outq + (size_t)mi * BNC;

        const float* Waw  = P(base + 0),  *Woff = P(base + 1);
        const float* Wout = P(base + 2),  *Wv   = P(base + 3);
        const float* baw  = P(base + 4),  *boff = P(base + 5);
        const float* bout = P(base + 6),  *bv   = P(base + 7);
        const float* W1   = P(base + 8),  *W2   = P(base + 9);
        const float* Wdw  = P(base + 10), *b1   = P(base + 11);
        const float* b2   = P(base + 12), *bdw  = P(base + 13);
        const float* fb   = P(base + 14), *fs   = P(base + 15);
        const float* qb   = P(base + 16), *qs   = P(base + 17);

        // --- deformable attention ---
        ln768_kernel<<<MROWS, 256, 0, stream>>>(qsrc, qs, qb, bufB);          // qn
        launch_gemm(stream, bufB, Woff, boff, nullptr, offb, MROWS, 48, CC, 1.f);
        launch_gemm(stream, bufB, Waw,  baw,  nullptr, awb,  MROWS, 24, CC, 1.f);
        launch_gemm(stream, featn, Wv,  bv,   nullptr, bufC, MROWS, CC, CC, 1.f);
        msda_sample_kernel<<<(BB * NN * HEADS) / 8, 256, 0, stream>>>(
            bufC, offb, awb, refp, bufB);                                     // attn_pre
        // q' = qsrc + 3 * (attn_pre @ Wout + bout)
        launch_gemm(stream, bufB, Wout, bout, qsrc, qdst, MROWS, CC, CC, 3.f);

        // --- conv FFN ---
        ln768_kernel<<<MROWS, 256, 0, stream>>>(qdst, fs, fb, bufC);
        launch_gemm(stream, bufC, W1, b1, nullptr, h1, MROWS, HID, CC, 1.f);
        dwconv_gelu_kernel<<<(MROWS * HID) / 256, 256, 0, stream>>>(h1, Wdw, bdw, hdw);
        // q'' = q' + (g @ W2 + b2)
        launch_gemm(stream, hdw, W2, b2, qdst, qdst, MROWS, CC, HID, 1.f);
    }
}